// TrajectoryFMHMT_10050223473173
// MI455X (gfx1250) — compile-verified
//
#include <hip/hip_runtime.h>
#include <hip/hip_bf16.h>

typedef __attribute__((ext_vector_type(2))) float v2f;
typedef __attribute__((ext_vector_type(4))) float v4f;
typedef __attribute__((ext_vector_type(8))) float v8f;

#define NB 16
#define NS 2048
#define NM 128
#define NC 32
#define ND 512
#define NT (NS + NM + NC)   // 2208
#define KNN1 17             // KNN + 1 (self included)

// ---------------- rank = cumsum(valid) - 1, per batch ----------------
__global__ void rank_kernel(const int* __restrict__ step_mask, int* __restrict__ rank) {
    int b = blockIdx.x;
    if (threadIdx.x == 0) {
        int acc = 0;
        const int* sm = step_mask + b * NS;
        int* rk = rank + b * NS;
        for (int i = 0; i < NS; ++i) {
            acc += (sm[i] > 0) ? 1 : 0;
            rk[i] = acc - 1;
        }
    }
}

// ---------------- graph base: diag | temporal | same (B128 NT stores) ----------------
__global__ void graph_base_kernel(const int* __restrict__ step_mask,
                                  const int* __restrict__ rank,
                                  const int* __restrict__ tokens,
                                  float* __restrict__ graph) {
    int i = blockIdx.x;
    int b = blockIdx.y;
    const int* sm = step_mask + b * NS;
    const int* rk = rank + b * NS;
    const int* tk = tokens + b * NS;
    bool vi = sm[i] > 0;
    int ri = rk[i];
    int ti = tk[i];
    float* g = graph + (size_t)b * NS * NS + (size_t)i * NS;
    // 256 threads x 4 consecutive floats = 1024 per sweep; 2 sweeps cover 2048
    for (int j4 = threadIdx.x * 4; j4 < NS; j4 += blockDim.x * 4) {
        v4f out;
#pragma unroll
        for (int u = 0; u < 4; ++u) {
            int j = j4 + u;
            bool vj = sm[j] > 0;
            bool pv = vi && vj;
            int dr = ri - rk[j];
            if (dr < 0) dr = -dr;
            bool edge = (i == j && vi) ||
                        (pv && dr <= 2) ||
                        (pv && ti == tk[j]);
            out[u] = edge ? 1.0f : 0.0f;
        }
        __builtin_nontemporal_store(out, (v4f*)(g + j4));
    }
}

// ---------------- knn edges: one wave per row ----------------
__global__ void knn_kernel(const float* __restrict__ coords,
                           const int* __restrict__ step_mask,
                           float* __restrict__ graph) {
    const int lane = threadIdx.x & 31;
    const int wv = threadIdx.x >> 5;
    const int i = blockIdx.x * (blockDim.x >> 5) + wv;
    const int b = blockIdx.y;
    if (i >= NS) return;

    const int* sm = step_mask + b * NS;
    if (!(sm[i] > 0)) return;  // invalid row: no finite distances, no edges

    const float* cb = coords + (size_t)b * NS * 2;
    const float xi = cb[i * 2 + 0];
    const float yi = cb[i * 2 + 1];
    const float si = xi * xi + yi * yi;
    const float INF = __builtin_inff();

    float best[KNN1];
#pragma unroll
    for (int t = 0; t < KNN1; ++t) best[t] = INF;

    // pass 1: per-lane sorted top-17 over strided candidates
    for (int j = lane; j < NS; j += 32) {
        float xj = cb[j * 2 + 0];
        float yj = cb[j * 2 + 1];
        float d2;
        if (sm[j] > 0) {
            float sj = xj * xj + yj * yj;
            d2 = si + sj - 2.0f * (xi * xj + yi * yj);
            d2 = fmaxf(d2, 0.0f);
        } else {
            d2 = INF;
        }
        if (d2 < best[KNN1 - 1]) {
            float v = d2;
#pragma unroll
            for (int t = 0; t < KNN1; ++t) {
                float lo = fminf(v, best[t]);
                float hi = fmaxf(v, best[t]);
                best[t] = lo;
                v = hi;
            }
        }
    }

    // merge across 32 lanes: 17 rounds of wave-min, consume winner's head
    float thr = INF;
#pragma unroll 1
    for (int t = 0; t < KNN1; ++t) {
        float v = best[0];
        float m = v;
#pragma unroll
        for (int off = 16; off >= 1; off >>= 1)
            m = fminf(m, __shfl_xor(m, off, 32));
        thr = m;
        unsigned long long ball = __ballot(v == m);
        int src = (int)__builtin_ctzll(ball);
        if (lane == src) {
#pragma unroll
            for (int q = 0; q < KNN1 - 1; ++q) best[q] = best[q + 1];
            best[KNN1 - 1] = INF;
        }
    }

    // pass 2: mark edges (i,j) and (j,i)
    float* gb = graph + (size_t)b * NS * NS;
    for (int j = lane; j < NS; j += 32) {
        if (!(sm[j] > 0)) continue;
        float xj = cb[j * 2 + 0];
        float yj = cb[j * 2 + 1];
        float sj = xj * xj + yj * yj;
        float d2 = fmaxf(si + sj - 2.0f * (xi * xj + yi * yj), 0.0f);
        if (d2 <= thr) {
            gb[(size_t)i * NS + j] = 1.0f;
            gb[(size_t)j * NS + i] = 1.0f;
        }
    }
}

// ---------------- attention mask (4 cols / thread, B128 NT stores) ----------------
__device__ __forceinline__ bool attn_val(int r, int c,
                                         const int* sm, const int* m2, const int* mk,
                                         const int* c2, const int* ck) {
    bool val;
    if (r < NS) {
        bool vr = sm[r] > 0;
        if (c < NS) {
            val = vr && (sm[c] > 0);
        } else if (c < NS + NM) {
            int mi = c - NS;
            int t = m2[r];
            val = vr && t >= 0 && t < NM && (mk[t] > 0) && (t == mi);
        } else {
            int ci = c - NS - NM;
            int t = c2[r];
            val = vr && t >= 0 && t < NC && (ck[t] > 0) && (t == ci);
        }
    } else if (r < NS + NM) {
        int mi = r - NS;
        if (c < NS) {
            int t = m2[c];
            val = (sm[c] > 0) && t >= 0 && t < NM && (mk[t] > 0) && (t == mi);
        } else if (c < NS + NM) {
            val = (r == c) && (mk[mi] > 0);
        } else {
            val = false;
        }
    } else {
        int ci = r - NS - NM;
        if (c < NS) {
            int t = c2[c];
            val = (sm[c] > 0) && t >= 0 && t < NC && (ck[t] > 0) && (t == ci);
        } else if (c < NS + NM) {
            val = false;
        } else {
            val = (r == c) && (ck[ci] > 0);
        }
    }
    return val || (r == c);
}

__global__ void attn_kernel(const int* __restrict__ step_mask,
                            const int* __restrict__ s2m,
                            const int* __restrict__ mmask,
                            const int* __restrict__ s2c,
                            const int* __restrict__ cmask,
                            float* __restrict__ attn) {
    const int b = blockIdx.y;
    const int TQ = NT / 4;  // 552 quads per row
    long long e4 = (long long)blockIdx.x * blockDim.x + threadIdx.x;
    if (e4 >= (long long)NT * TQ) return;
    int r = (int)(e4 / TQ);
    int c4 = (int)(e4 % TQ) * 4;

    const int* sm = step_mask + b * NS;
    const int* m2 = s2m + b * NS;
    const int* mk = mmask + b * NM;
    const int* c2 = s2c + b * NS;
    const int* ck = cmask + b * NC;

    v4f out;
#pragma unroll
    for (int u = 0; u < 4; ++u)
        out[u] = attn_val(r, c4 + u, sm, m2, mk, c2, ck) ? 1.0f : 0.0f;

    float* dst = attn + (size_t)b * NT * NT + (size_t)r * NT + c4;
    __builtin_nontemporal_store(out, (v4f*)dst);
}

// ---------------- proj = H(MxD) * W^T(DxD) via V_WMMA_F32_16X16X4_F32 ----------------
// proj[b,m,e] = sum_d H[b,m,d] * W[e,d]   (B operand column n=e loads W[e,k..k+1])
__global__ void proj_wmma_kernel(const float* __restrict__ H,
                                 const float* __restrict__ W,
                                 float* __restrict__ P) {
    const int lane = threadIdx.x & 31;
    const int e0 = blockIdx.x * 16;
    const int m0 = blockIdx.y * 16;
    const int b = blockIdx.z;
    const int half = lane >> 4;
    const int l16 = lane & 15;

    const float* Hb = H + (size_t)b * NM * ND;
    v8f c = {};
    for (int k = 0; k < ND; k += 4) {
        int ka = k + half * 2;
        const float* ap = Hb + (size_t)(m0 + l16) * ND + ka;
        const float* bp = W + (size_t)(e0 + l16) * ND + ka;
        v2f a, bb;
        a.x = ap[0]; a.y = ap[1];
        bb.x = bp[0]; bb.y = bp[1];
        c = __builtin_amdgcn_wmma_f32_16x16x4_f32(false, a, false, bb,
                                                  (short)0, c, false, false);
    }
    float* Pb = P + (size_t)b * NM * ND;
#pragma unroll
    for (int r = 0; r < 8; ++r) {
        Pb[(size_t)(m0 + r + 8 * half) * ND + e0 + l16] = c[r];
    }
}

// ---------------- logits = (P * P^T) / sqrt(D), masked ----------------
__global__ void logits_wmma_kernel(const float* __restrict__ P,
                                   const int* __restrict__ mmask,
                                   float* __restrict__ L) {
    const int lane = threadIdx.x & 31;
    const int n0 = blockIdx.x * 16;
    const int m0 = blockIdx.y * 16;
    const int b = blockIdx.z;
    const int half = lane >> 4;
    const int l16 = lane & 15;

    const float* Pb = P + (size_t)b * NM * ND;
    v8f c = {};
    for (int k = 0; k < ND; k += 4) {
        int ka = k + half * 2;
        const float* ap = Pb + (size_t)(m0 + l16) * ND + ka;
        const float* bp = Pb + (size_t)(n0 + l16) * ND + ka;  // B[k,n] = P[n,k]
        v2f a, bb;
        a.x = ap[0]; a.y = ap[1];
        bb.x = bp[0]; bb.y = bp[1];
        c = __builtin_amdgcn_wmma_f32_16x16x4_f32(false, a, false, bb,
                                                  (short)0, c, false, false);
    }
    const float scale = 0.044194173824159216f;  // 1/sqrt(512)
    const int* mk = mmask + b * NM;
    bool mn = mk[n0 + l16] > 0;
    float* Lb = L + (size_t)b * NM * NM;
#pragma unroll
    for (int r = 0; r < 8; ++r) {
        int row = m0 + r + 8 * half;
        bool mm = mk[row] > 0;
        Lb[(size_t)row * NM + n0 + l16] = (mm && mn) ? c[r] * scale : 0.0f;
    }
}

extern "C" void kernel_launch(void* const* d_in, const int* in_sizes, int n_in,
                              void* d_out, int out_size, void* d_ws, size_t ws_size,
                              hipStream_t stream) {
    const float* coords     = (const float*)d_in[0];
    const int*   tokens     = (const int*)d_in[1];
    const int*   step_mask  = (const int*)d_in[2];
    const int*   s2m        = (const int*)d_in[3];
    const int*   mmask      = (const int*)d_in[4];
    const int*   s2c        = (const int*)d_in[5];
    const int*   cmask      = (const int*)d_in[6];
    const float* mid_hidden = (const float*)d_in[7];
    const float* W_flow     = (const float*)d_in[8];

    float* graph  = (float*)d_out;
    float* attn   = graph + (size_t)NB * NS * NS;
    float* logits = attn + (size_t)NB * NT * NT;

    int*   rank = (int*)d_ws;                              // NB*NS ints   (128 KB)
    float* proj = (float*)((char*)d_ws + (size_t)NB * NS * sizeof(int));  // NB*NM*ND f32 (4 MB)

    rank_kernel<<<NB, 32, 0, stream>>>(step_mask, rank);
    graph_base_kernel<<<dim3(NS, NB), 256, 0, stream>>>(step_mask, rank, tokens, graph);
    knn_kernel<<<dim3(NS / 8, NB), 256, 0, stream>>>(coords, step_mask, graph);

    long long quads = (long long)NT * (NT / 4);  // 2208 * 552
    attn_kernel<<<dim3((unsigned)((quads + 255) / 256), NB), 256, 0, stream>>>(
        step_mask, s2m, mmask, s2c, cmask, attn);

    proj_wmma_kernel<<<dim3(ND / 16, NM / 16, NB), 32, 0, stream>>>(mid_hidden, W_flow, proj);
    logits_wmma_kernel<<<dim3(NM / 16, NM / 16, NB), 32, 0, stream>>>(proj, mmask, logits);
}